// RetinaNetLosses_65678639890458
// MI455X (gfx1250) — compile-verified
//
#include <hip/hip_runtime.h>
#include <math.h>
#include <stdint.h>

// ---------------- problem constants ----------------
#define B_      8
#define A_      120000
#define M_      64
#define C_      80
#define THREADS 256
#define SUBANCH 64                    // anchors per async sub-tile
#define SUBDW   (SUBANCH * C_)        // 5120 dwords per sub-tile
#define SUBF4   (SUBDW / 4)           // 1280 float4 per sub-tile
#define F4PT    (SUBF4 / THREADS)     // 5 float4 (b128) issues per thread per sub-tile
#define NSUB    4                     // sub-tiles per block => 256 anchors per block
#define NBLK    ((A_ + THREADS - 1) / THREADS)   // 469 blocks per image

// CDNA5 async global -> LDS copy (ASYNCcnt path), GVS addressing:
//   lds[vdst_lds + 0..15] = mem[saddr + voffset + 0..15]
static __device__ __forceinline__ void async_b128(uint32_t lds_off, uint32_t voff,
                                                  const void* sbase) {
  asm volatile("global_load_async_to_lds_b128 %0, %1, %2"
               :: "v"(lds_off), "v"(voff), "s"(sbase)
               : "memory");
}

__global__ __launch_bounds__(THREADS)
void retina_partial(const float* __restrict__ cls,  // [B,A,C]
                    const float* __restrict__ bbp,  // [B,A,4]
                    const float* __restrict__ anc,  // [B,A,4] ccwh
                    const float* __restrict__ box,  // [B,M,4] xyxy
                    const int*   __restrict__ lab,  // [B,M]
                    float*       __restrict__ ws)   // [B, NBLK, 3]
{
  __shared__ __align__(16) float  s_cls[2][SUBDW];   // double-buffered logits (40 KB)
  __shared__ __align__(16) float4 s_box[M_];
  __shared__ int   s_lab[M_];
  __shared__ int   s_tlbl[THREADS];
  __shared__ float s_mask[THREADS];

  const int tid = threadIdx.x;
  const int blk = blockIdx.x;
  const int b   = blockIdx.y;

  // ---- kick off async DMA for the first two logits sub-tiles immediately ----
  // (overlaps the whole IoU-matching phase with HBM streaming)
  const size_t base_dw  = ((size_t)b * A_ + (size_t)blk * THREADS) * C_;
  const size_t max_voff = (size_t)B_ * A_ * C_ * 4 - 16;   // clamp for tail block

  auto issue = [&](int s, int buf) {
    const uint32_t lbase = (uint32_t)(uintptr_t)(&s_cls[buf][0]);
#pragma unroll
    for (int k = 0; k < F4PT; ++k) {
      const int i4 = tid + THREADS * k;
      size_t voff = (base_dw + (size_t)s * SUBDW + (size_t)i4 * 4) * 4;
      if (voff > max_voff) voff = max_voff;
      async_b128(lbase + (uint32_t)(i4 * 16), (uint32_t)voff, (const void*)cls);
    }
  };
  issue(0, 0);
  issue(1, 1);                                       // 2 tiles (10 instr/wave) in flight

  // stage boxes + labels for this image into LDS
  if (tid < M_) {
    s_box[tid] = reinterpret_cast<const float4*>(box)[b * M_ + tid];
    s_lab[tid] = lab[b * M_ + tid];
  }
  __syncthreads();

  // ---------------- Phase A: anchor matching + smooth-L1 ----------------
  const int  a     = blk * THREADS + tid;
  const bool valid = (a < A_);
  const int  ac    = valid ? a : (A_ - 1);           // clamp for safe loads

  float4 av = reinterpret_cast<const float4*>(anc)[(size_t)b * A_ + ac];
  const float ax1 = av.x - 0.5f * av.z, ay1 = av.y - 0.5f * av.w;
  const float ax2 = av.x + 0.5f * av.z, ay2 = av.y + 0.5f * av.w;
  const float areaA = av.z * av.w;

  float best = -1.0f; int bidx = 0;
#pragma unroll 4
  for (int m = 0; m < M_; ++m) {
    float4 bv = s_box[m];                            // LDS broadcast read
    float ltx = fmaxf(ax1, bv.x), lty = fmaxf(ay1, bv.y);
    float rbx = fminf(ax2, bv.z), rby = fminf(ay2, bv.w);
    float w = fmaxf(rbx - ltx, 0.f), h = fmaxf(rby - lty, 0.f);
    float inter = w * h;
    float areaB = (bv.z - bv.x) * (bv.w - bv.y);
    float iou = inter / (areaA + areaB - inter);
    if (iou > best) { best = iou; bidx = m; }        // strict > == first-index argmax
  }
  const int  match = (best < 0.4f) ? -1 : ((best > 0.5f) ? bidx : -2);
  const bool pos   = valid && (match >= 0);
  const float np_acc = pos ? 1.f : 0.f;

  // bbox_2_activ + smooth-L1 (beta=1), masked by pos
  float4 tb = s_box[(match >= 0) ? match : 0];
  float bcx = 0.5f * (tb.x + tb.z), bcy = 0.5f * (tb.y + tb.w);
  float bw  = tb.z - tb.x,          bh  = tb.w - tb.y;
  float t0 = (bcx - av.x) / av.z * 10.0f;            // / 0.1
  float t1 = (bcy - av.y) / av.w * 10.0f;
  float t2 = __logf(bw / av.z + 1e-8f) * 5.0f;       // / 0.2
  float t3 = __logf(bh / av.w + 1e-8f) * 5.0f;
  float4 pv = reinterpret_cast<const float4*>(bbp)[(size_t)b * A_ + ac];
  float bb_acc = 0.f;
  {
    float d;
    d = fabsf(pv.x - t0); bb_acc += (d < 1.f) ? 0.5f * d * d : (d - 0.5f);
    d = fabsf(pv.y - t1); bb_acc += (d < 1.f) ? 0.5f * d * d : (d - 0.5f);
    d = fabsf(pv.z - t2); bb_acc += (d < 1.f) ? 0.5f * d * d : (d - 0.5f);
    d = fabsf(pv.w - t3); bb_acc += (d < 1.f) ? 0.5f * d * d : (d - 0.5f);
    bb_acc = pos ? bb_acc : 0.f;
  }

  s_tlbl[tid] = (match >= 0) ? s_lab[match] : 0;     // 0 = background
  s_mask[tid] = (valid && match >= -1) ? 1.f : 0.f;  // -2 is ignored

  // ---------------- Phase B: focal loss, async double-buffered streaming ----------------
  float cls_acc = 0.f;
  for (int s = 0; s < NSUB; ++s) {
    const int cur = s & 1;
    if (s + 1 < NSUB) {
      // one newer tile still in flight; loads retire in order -> oldest tile done
      asm volatile("s_wait_asynccnt %0" :: "n"(F4PT) : "memory");
    } else {
      asm volatile("s_wait_asynccnt 0" ::: "memory");
    }
    __syncthreads();                                 // buffer + tlbl/mask visible block-wide

#pragma unroll
    for (int k = 0; k < F4PT; ++k) {
      const int i4 = tid + THREADS * k;              // stride-1 across lanes: no conflicts
      const int la = s * SUBANCH + i4 / 20;          // 20 float4 per anchor (80/4)
      const float mk = s_mask[la];
      const int   tl = s_tlbl[la];
      float4 v = *reinterpret_cast<const float4*>(&s_cls[cur][i4 * 4]);
      float e[4] = {v.x, v.y, v.z, v.w};
      const int cbase = (i4 % 20) * 4 + 1;           // 1-based class id
#pragma unroll
      for (int j = 0; j < 4; ++j) {
        const float p = e[j];
        const float t = (tl == cbase + j) ? 1.f : 0.f;
        // shared transcendentals: z = exp(-|p|), r = sigmoid(|p|)
        const float z  = __expf(-fabsf(p));
        const float r  = 1.f / (1.f + z);
        const float ps = (p >= 0.f) ? r : (1.f - r); // sigmoid(p)
        float w  = (t != 0.f) ? (1.f - ps) : ps;     // t*(1-ps)+(1-t)*ps
        const float al = (t != 0.f) ? 0.75f : 0.25f; // 1-ALPHA / ALPHA
        w = w * w * al;                              // GAMMA = 2
        const float bce = fmaxf(p, 0.f) - p * t + __logf(1.f + z); // softplus(-|p|)
        cls_acc += w * bce * mk;
      }
    }
    __syncthreads();                                 // everyone done reading buf[cur]
    if (s + 2 < NSUB) issue(s + 2, cur);             // refill the buffer just freed
  }

  // ---------------- block reduction -> 3 partials ----------------
  float* red = &s_cls[0][0];
  red[tid]               = cls_acc;
  red[THREADS + tid]     = bb_acc;
  red[2 * THREADS + tid] = np_acc;
  __syncthreads();
  for (int st = THREADS / 2; st > 0; st >>= 1) {
    if (tid < st) {
      red[tid]               += red[tid + st];
      red[THREADS + tid]     += red[THREADS + tid + st];
      red[2 * THREADS + tid] += red[2 * THREADS + tid + st];
    }
    __syncthreads();
  }
  if (tid == 0) {
    float* o = ws + ((size_t)b * gridDim.x + blk) * 3;
    o[0] = red[0];
    o[1] = red[THREADS];
    o[2] = red[2 * THREADS];
  }
}

__global__ __launch_bounds__(THREADS)
void retina_reduce(const float* __restrict__ ws, float* __restrict__ out, int nblk) {
  __shared__ float r0[THREADS], r1[THREADS], r2[THREADS];
  const int tid = threadIdx.x;
  float cm = 0.f, bm = 0.f;
  for (int b = 0; b < B_; ++b) {
    float c = 0.f, r = 0.f, n = 0.f;
    for (int i = tid; i < nblk; i += THREADS) {
      const float* p = ws + ((size_t)b * nblk + i) * 3;
      c += p[0]; r += p[1]; n += p[2];
    }
    r0[tid] = c; r1[tid] = r; r2[tid] = n;
    __syncthreads();
    for (int st = THREADS / 2; st > 0; st >>= 1) {
      if (tid < st) { r0[tid] += r0[tid + st]; r1[tid] += r1[tid + st]; r2[tid] += r2[tid + st]; }
      __syncthreads();
    }
    if (tid == 0) {
      const float np = r2[0];
      cm += r0[0] / fmaxf(np, 1.f);
      bm += r1[0] / fmaxf(4.f * np, 1.f);
    }
    __syncthreads();
  }
  if (tid == 0) {
    out[0] = cm * (1.f / B_);
    out[1] = bm * (1.f / B_);
  }
}

extern "C" void kernel_launch(void* const* d_in, const int* in_sizes, int n_in,
                              void* d_out, int out_size, void* d_ws, size_t ws_size,
                              hipStream_t stream) {
  (void)in_sizes; (void)n_in; (void)out_size; (void)ws_size;
  const float* cls = (const float*)d_in[0];
  const float* bbp = (const float*)d_in[1];
  const float* anc = (const float*)d_in[2];
  const float* box = (const float*)d_in[3];
  const int*   lab = (const int*)d_in[4];
  float* ws = (float*)d_ws;   // NBLK*B_*3 floats = ~45 KB of scratch

  dim3 grid(NBLK, B_);
  retina_partial<<<grid, THREADS, 0, stream>>>(cls, bbp, anc, box, lab, ws);
  retina_reduce<<<1, THREADS, 0, stream>>>(ws, (float*)d_out, NBLK);
}